// BasicConv_7112465842256
// MI455X (gfx1250) — compile-verified
//
#include <hip/hip_runtime.h>

typedef __attribute__((ext_vector_type(2))) float v2f;
typedef __attribute__((ext_vector_type(8))) float v8f;
typedef __attribute__((ext_vector_type(4))) unsigned int u32x4;
typedef __attribute__((ext_vector_type(8))) int i32x8;
typedef __attribute__((ext_vector_type(4))) int i32x4;

#define B_    16
#define C_    128
#define HALF_ 64
#define H_    128
#define W_    128
#define HW_   16384
#define EPS_  1e-5f

// ---------------- depthwise 3x3 on even channels -> T1 (ws) ----------------
// x is streamed once (no long-range reuse): use non-temporal loads so the
// 192 MB L2 stays reserved for T1 + the fusion tensor.
__global__ __launch_bounds__(256) void dw_kernel(const float* __restrict__ x,
                                                 const float* __restrict__ dw_w,
                                                 const float* __restrict__ dw_b,
                                                 float* __restrict__ t1)
{
    int tid = blockIdx.x * 256 + threadIdx.x;       // over B*64*HW = 16.7M
    int s = tid & (HW_ - 1);
    int c = (tid >> 14) & (HALF_ - 1);
    int b = tid >> 20;
    int h = s >> 7, w = s & (W_ - 1);
    const float* xp = x + ((size_t)b * C_ + 2 * c) * HW_;
    const float* wp = dw_w + c * 9;
    float acc = dw_b[c];
#pragma unroll
    for (int i = 0; i < 3; ++i) {
        int hh = h + i - 1;
        if ((unsigned)hh < (unsigned)H_) {
#pragma unroll
            for (int j = 0; j < 3; ++j) {
                int ww = w + j - 1;
                if ((unsigned)ww < (unsigned)W_)
                    acc = fmaf(wp[i * 3 + j],
                               __builtin_nontemporal_load(&xp[hh * W_ + ww]), acc);
            }
        }
    }
    t1[tid] = acc;
}

// ---------------- pointwise 64x64 GEMM: TDM -> LDS -> V_WMMA_F32_16X16X4 ---
// Out[b, 0:64, :] = pw_w(64x64) * T1[b](64x16384) + pw_b
// One TENSOR_LOAD_TO_LDS per block stages the 64(K) x 128(N) B tile (32 KB);
// 8 waves each compute a 64(M) x 16(N) tile = 64 wmma ops.
__global__ __launch_bounds__(256) void pw_wmma_kernel(const float* __restrict__ t1,
                                                      const float* __restrict__ pw_w,
                                                      const float* __restrict__ pw_b,
                                                      float* __restrict__ out)
{
    __shared__ float wsh[HALF_ * HALF_];            // A: [oc][ic], 16 KB
    __shared__ float bsh[HALF_ * 128];              // B tile: [k][col], 32 KB

    int lane = threadIdx.x & 31;
    int wave = threadIdx.x >> 5;
    int b      = blockIdx.x >> 7;                   // 128 blocks per batch image
    int colblk = blockIdx.x & 127;                  // 128 columns per block

    for (int i = threadIdx.x; i < HALF_ * HALF_; i += 256) wsh[i] = pw_w[i];

    if (wave == 0) {
        // Tensor DMA descriptor (ISA 8.3/8.4): 2D tile, 4-byte elements.
        unsigned long long ga =
            (unsigned long long)(uintptr_t)(t1 + (size_t)b * HALF_ * HW_ + colblk * 128);
        unsigned int lds_off = (unsigned int)(uintptr_t)(&bsh[0]); // flat->LDS: low 32 bits
        u32x4 g0;
        g0[0] = 1u;                                                 // count=1, user D#
        g0[1] = lds_off;                                            // lds_addr (bytes)
        g0[2] = (unsigned int)ga;                                   // global_addr[31:0]
        g0[3] = (unsigned int)((ga >> 32) & 0x01FFFFFFu) | 0x80000000u; // [56:32] | type=2
        i32x8 g1;
        g1[0] = 0x00020000;                         // workgroup_mask=0, data_size=2 (4B)
        g1[1] = (int)((HW_ & 0xFFFF) << 16);        // tensor_dim0[15:0] = 16384
        g1[2] = (int)((HW_ >> 16) | (HALF_ << 16)); // tensor_dim0[31:16] | tensor_dim1[15:0]
        g1[3] = (int)(128 << 16);                   // tensor_dim1[31:16] | tile_dim0=128
        g1[4] = HALF_;                              // tile_dim1=64, tile_dim2=0
        g1[5] = HW_;                                // tensor_dim0_stride[31:0] = 16384
        g1[6] = 0;                                  // stride hi | tensor_dim1_stride lo
        g1[7] = 0;
        i32x4 z4 = {0, 0, 0, 0};                    // groups 2/3 unused (2-D tensor)
        i32x8 z8 = {0, 0, 0, 0, 0, 0, 0, 0};
        __builtin_amdgcn_tensor_load_to_lds(g0, g1, z4, z4, z8, 0);
        __builtin_amdgcn_s_wait_tensorcnt(0);
    }
    __syncthreads();

    int nl = lane & 15;                             // M / N index within 16x16 tile
    int kh = lane >> 4;                             // K-half select (ISA f32 A/B layout)
    int ncol = wave * 16 + nl;                      // column within block tile

    v8f acc0 = {}, acc1 = {}, acc2 = {}, acc3 = {};

    for (int k0 = 0; k0 < HALF_; k0 += 4) {
        int ka = k0 + kh * 2;
        v2f bf;                                     // B fragment (4x16) from LDS tile
        bf.x = bsh[ka * 128 + ncol];
        bf.y = bsh[(ka + 1) * 128 + ncol];
        v2f a0, a1, a2, a3;                         // A fragments for M tiles 0,16,32,48
        a0.x = wsh[(nl +  0) * HALF_ + ka]; a0.y = wsh[(nl +  0) * HALF_ + ka + 1];
        a1.x = wsh[(nl + 16) * HALF_ + ka]; a1.y = wsh[(nl + 16) * HALF_ + ka + 1];
        a2.x = wsh[(nl + 32) * HALF_ + ka]; a2.y = wsh[(nl + 32) * HALF_ + ka + 1];
        a3.x = wsh[(nl + 48) * HALF_ + ka]; a3.y = wsh[(nl + 48) * HALF_ + ka + 1];

        acc0 = __builtin_amdgcn_wmma_f32_16x16x4_f32(false, a0, false, bf, (short)0, acc0, false, false);
        acc1 = __builtin_amdgcn_wmma_f32_16x16x4_f32(false, a1, false, bf, (short)0, acc1, false, false);
        acc2 = __builtin_amdgcn_wmma_f32_16x16x4_f32(false, a2, false, bf, (short)0, acc2, false, false);
        acc3 = __builtin_amdgcn_wmma_f32_16x16x4_f32(false, a3, false, bf, (short)0, acc3, false, false);
    }

    // D layout: VGPR r holds rows r (lanes 0-15) and r+8 (lanes 16-31)
    float* outb = out + (size_t)b * C_ * HW_;       // channels 0..63
    int col = colblk * 128 + ncol;
#pragma unroll
    for (int r = 0; r < 8; ++r) {
        int m = r + 8 * kh;
        outb[(size_t)(m +  0) * HW_ + col] = acc0[r] + pw_b[m +  0];
        outb[(size_t)(m + 16) * HW_ + col] = acc1[r] + pw_b[m + 16];
        outb[(size_t)(m + 32) * HW_ + col] = acc2[r] + pw_b[m + 32];
        outb[(size_t)(m + 48) * HW_ + col] = acc3[r] + pw_b[m + 48];
    }
}

// ---------------- dilated depthwise on odd channels -> out[:,64:128] -------
// Each 256-thread block sits in a single (b, c) plane, so BN partial stats
// are fused here: one LDS tree + 2 global atomics per block.
__global__ __launch_bounds__(256) void mcc_kernel(const float* __restrict__ x,
                                                  const float* __restrict__ mcc_w,
                                                  const float* __restrict__ mcc_b,
                                                  float* __restrict__ out,
                                                  float* __restrict__ stats)
{
    __shared__ float ssum[256];
    __shared__ float ssq[256];
    int tid = blockIdx.x * 256 + threadIdx.x;       // over B*64*HW
    int s = tid & (HW_ - 1);
    int c = (tid >> 14) & (HALF_ - 1);              // uniform per block
    int b = tid >> 20;
    int h = s >> 7, w = s & (W_ - 1);
    int j = c & 3;
    int d = j + 1;
    const float* xp = x + ((size_t)b * C_ + 2 * c + 1) * HW_;
    const float* wp = mcc_w + j * 9;
    float acc = mcc_b[j];
#pragma unroll
    for (int i = 0; i < 3; ++i) {
        int hh = h + (i - 1) * d;
        if ((unsigned)hh < (unsigned)H_) {
#pragma unroll
            for (int jj = 0; jj < 3; ++jj) {
                int ww = w + (jj - 1) * d;
                if ((unsigned)ww < (unsigned)W_)
                    acc = fmaf(wp[i * 3 + jj],
                               __builtin_nontemporal_load(&xp[hh * W_ + ww]), acc);
            }
        }
    }
    out[((size_t)b * C_ + HALF_ + c) * HW_ + s] = acc;

    ssum[threadIdx.x] = acc;
    ssq[threadIdx.x] = acc * acc;
    __syncthreads();
    for (int off = 128; off > 0; off >>= 1) {
        if ((int)threadIdx.x < off) {
            ssum[threadIdx.x] += ssum[threadIdx.x + off];
            ssq[threadIdx.x]  += ssq[threadIdx.x + off];
        }
        __syncthreads();
    }
    if (threadIdx.x == 0) {
        atomicAdd(&stats[HALF_ + c], ssum[0]);
        atomicAdd(&stats[C_ + HALF_ + c], ssq[0]);
    }
}

// ---------------- batchnorm statistics (pw half only, L2-resident) ---------
__global__ __launch_bounds__(256) void stats_zero(float* __restrict__ stats)
{
    stats[threadIdx.x] = 0.0f;                      // sum[128], sumsq[128]
}

__global__ __launch_bounds__(256) void stats_kernel(const float* __restrict__ out,
                                                    float* __restrict__ stats)
{
    __shared__ float ssum[256];
    __shared__ float ssq[256];
    int p = blockIdx.x >> 2;                        // p = b*64 + c, c in [0,64)
    int chunk = blockIdx.x & 3;                     // 4 blocks per plane
    int b = p >> 6;
    int c = p & (HALF_ - 1);
    const float* base = out + ((size_t)b * C_ + c) * HW_ + chunk * 4096;
    float s = 0.0f, q = 0.0f;
#pragma unroll
    for (int i = 0; i < 16; ++i) {
        float v = base[threadIdx.x + i * 256];
        s += v;
        q = fmaf(v, v, q);
    }
    ssum[threadIdx.x] = s;
    ssq[threadIdx.x] = q;
    __syncthreads();
    for (int off = 128; off > 0; off >>= 1) {
        if ((int)threadIdx.x < off) {
            ssum[threadIdx.x] += ssum[threadIdx.x + off];
            ssq[threadIdx.x]  += ssq[threadIdx.x + off];
        }
        __syncthreads();
    }
    if (threadIdx.x == 0) {
        atomicAdd(&stats[c], ssum[0]);
        atomicAdd(&stats[C_ + c], ssq[0]);
    }
}

// ---------------- normalize + affine + relu, in place, float4 --------------
__global__ __launch_bounds__(256) void norm_kernel(const float* __restrict__ stats,
                                                   const float* __restrict__ gamma,
                                                   const float* __restrict__ beta,
                                                   float* __restrict__ out)
{
    int vid = blockIdx.x * 256 + threadIdx.x;       // over 33.5M/4 float4s
    size_t idx = (size_t)vid * 4;
    int c = (int)((idx >> 14) & (C_ - 1));          // HW_ % 4 == 0 -> uniform c
    const float inv_n = 1.0f / ((float)B_ * (float)HW_);
    float mean = stats[c] * inv_n;
    float var  = stats[C_ + c] * inv_n - mean * mean;
    float rs = rsqrtf(var + EPS_);
    float g = gamma[c] * rs;
    float bb = fmaf(-mean, g, beta[c]);
    float4 v = ((const float4*)out)[vid];
    v.x = fmaxf(fmaf(v.x, g, bb), 0.0f);
    v.y = fmaxf(fmaf(v.y, g, bb), 0.0f);
    v.z = fmaxf(fmaf(v.z, g, bb), 0.0f);
    v.w = fmaxf(fmaf(v.w, g, bb), 0.0f);
    ((float4*)out)[vid] = v;
}

extern "C" void kernel_launch(void* const* d_in, const int* in_sizes, int n_in,
                              void* d_out, int out_size, void* d_ws, size_t ws_size,
                              hipStream_t stream)
{
    const float* x     = (const float*)d_in[0];
    const float* dw_w  = (const float*)d_in[1];
    const float* dw_b  = (const float*)d_in[2];
    const float* pw_w  = (const float*)d_in[3];
    const float* pw_b  = (const float*)d_in[4];
    const float* mcc_w = (const float*)d_in[5];
    const float* mcc_b = (const float*)d_in[6];
    const float* gamma = (const float*)d_in[7];
    const float* beta  = (const float*)d_in[8];
    float* out = (float*)d_out;

    float* t1    = (float*)d_ws;                    // B*64*HW floats = 67 MB
    float* stats = (float*)((char*)d_ws + (size_t)B_ * HALF_ * HW_ * sizeof(float));

    stats_zero     <<<1,                          256, 0, stream>>>(stats);
    dw_kernel      <<<(B_ * HALF_ * HW_) / 256,   256, 0, stream>>>(x, dw_w, dw_b, t1);
    pw_wmma_kernel <<<B_ * 128,                   256, 0, stream>>>(t1, pw_w, pw_b, out);
    mcc_kernel     <<<(B_ * HALF_ * HW_) / 256,   256, 0, stream>>>(x, mcc_w, mcc_b, out, stats);
    stats_kernel   <<<B_ * HALF_ * 4,             256, 0, stream>>>(out, stats);
    norm_kernel    <<<(B_ * C_ * HW_) / (256 * 4), 256, 0, stream>>>(stats, gamma, beta, out);
}